// DigitCaps_10505490006076
// MI455X (gfx1250) — compile-verified
//
#include <hip/hip_runtime.h>
#include <hip/hip_bf16.h>

// DigitCaps dynamic routing, fully fused for MI455X (gfx1250).
// u_hat ([10,512,1152,16] f32 = 377 MB) is NEVER materialized to HBM: it is
// recomputed per routing pass with V_WMMA_F32_16X16X4_F32 from L2-resident
// x (19 MB) and W (6 MB). Routing state (logits / coupling coeffs) lives in LDS.

typedef float v2f __attribute__((ext_vector_type(2)));
typedef float v8f __attribute__((ext_vector_type(8)));

#define Bq    512
#define Pq    1152
#define CIq   8
#define Nq    10
#define COq   16
#define LSTR  1153          // padded stride for [16][P] LDS arrays (bank-conflict free)
#define SSTR  17            // padded stride for 16x16 tiles
#define NTHREADS 512
#define NWAVES   (NTHREADS / 32)
#define PPW      (Pq / NWAVES)   // 72 p-values per wave

// Compute the 16(b) x 16(co) u_hat tile for one p via two f32 WMMAs (K=4+4).
// ISA 16x4 f32 A layout: lanes 0-15 -> M=lane, VGPR0=K0,VGPR1=K1;
//                        lanes 16-31 -> M=lane-16, VGPR0=K2,VGPR1=K3.
// B (4x16) mirrors: VGPR0/1 hold rows {2*half, 2*half+1}, N = lane&15.
__device__ __forceinline__ v8f uhat_tile(const float* __restrict__ xp,
                                         const float* __restrict__ wp,
                                         int hh, int mrow)
{
    v2f a0 = *(const v2f*)(xp + 2 * hh);        // x[b, p, 2h..2h+1]
    v2f a1 = *(const v2f*)(xp + 4 + 2 * hh);    // x[b, p, 4+2h..4+2h+1]
    v2f b0, b1;
    b0.x = wp[(2 * hh + 0) * COq + mrow];       // W[n, p, 2h,   o]
    b0.y = wp[(2 * hh + 1) * COq + mrow];       // W[n, p, 2h+1, o]
    b1.x = wp[(2 * hh + 4) * COq + mrow];       // W[n, p, 4+2h, o]
    b1.y = wp[(2 * hh + 5) * COq + mrow];       // W[n, p, 5+2h, o]
    v8f u = {0.f, 0.f, 0.f, 0.f, 0.f, 0.f, 0.f, 0.f};
    u = __builtin_amdgcn_wmma_f32_16x16x4_f32(false, a0, false, b0, (short)0, u, false, false);
    u = __builtin_amdgcn_wmma_f32_16x16x4_f32(false, a1, false, b1, (short)0, u, false, false);
    return u;  // C/D layout: lane L, vgpr r -> M = r + 8*(L>=16), N = L&15
}

// Agreement pass: logits[b][p] (+)= sum_o u_hat[b,p,o] * v[b,o]
__device__ __forceinline__ void a_pass(const float* __restrict__ xrow,
                                       const float* __restrict__ wbase,
                                       int p_begin, int p_end, int hh, int mrow,
                                       const float* __restrict__ v_lds,
                                       float* __restrict__ logits, bool accum)
{
    float vv[8];
#pragma unroll
    for (int r = 0; r < 8; ++r) vv[r] = v_lds[(r + 8 * hh) * SSTR + mrow];

    for (int p = p_begin; p < p_end; ++p) {
        v8f u = uhat_tile(xrow + (size_t)p * CIq, wbase + (size_t)p * CIq * COq, hh, mrow);
#pragma unroll
        for (int r = 0; r < 8; ++r) {
            float prod = u[r] * vv[r];
            // sum over the 16 lanes (o dimension) of this half-wave
            prod += __shfl_xor(prod, 1);
            prod += __shfl_xor(prod, 2);
            prod += __shfl_xor(prod, 4);
            prod += __shfl_xor(prod, 8);
            if (mrow == 0) {
                float* dst = &logits[(r + 8 * hh) * LSTR + p];
                if (accum) *dst += prod; else *dst = prod;
            }
        }
    }
}

// Weighted-sum pass: s[b][o] += sum_p c[b][p] * u_hat[b,p,o]
__device__ __forceinline__ void s_pass(const float* __restrict__ xrow,
                                       const float* __restrict__ wbase,
                                       int p_begin, int p_end, int hh, int mrow,
                                       const float* __restrict__ cw,
                                       float* __restrict__ s_red, bool uniform)
{
    v8f sacc = {0.f, 0.f, 0.f, 0.f, 0.f, 0.f, 0.f, 0.f};
    const float cu = 1.0f / (float)Pq;
    for (int p = p_begin; p < p_end; ++p) {
        v8f u = uhat_tile(xrow + (size_t)p * CIq, wbase + (size_t)p * CIq * COq, hh, mrow);
#pragma unroll
        for (int r = 0; r < 8; ++r) {
            float c = uniform ? cu : cw[(r + 8 * hh) * LSTR + p];  // LDS broadcast
            sacc[r] += c * u[r];
        }
    }
#pragma unroll
    for (int r = 0; r < 8; ++r)
        atomicAdd(&s_red[(r + 8 * hh) * SSTR + mrow], sacc[r]);   // ds_add_f32
}

// Softmax over p per b-row: c[b][p] = exp(logit-max)/sum.  16 threads per row.
__device__ __forceinline__ void stats_pass(int tid,
                                           const float* __restrict__ logits,
                                           float* __restrict__ cw)
{
    if (tid < 256) {
        int row = tid >> 4, sub = tid & 15;
        const float* lr = &logits[row * LSTR];
        float m = -3.4e38f;
        for (int p = sub; p < Pq; p += 16) m = fmaxf(m, lr[p]);
        m = fmaxf(m, __shfl_xor(m, 1));
        m = fmaxf(m, __shfl_xor(m, 2));
        m = fmaxf(m, __shfl_xor(m, 4));
        m = fmaxf(m, __shfl_xor(m, 8));
        float s = 0.f;
        for (int p = sub; p < Pq; p += 16) s += __expf(lr[p] - m);
        s += __shfl_xor(s, 1);
        s += __shfl_xor(s, 2);
        s += __shfl_xor(s, 4);
        s += __shfl_xor(s, 8);
        float inv = 1.0f / s;
        float* cr = &cw[row * LSTR];
        for (int p = sub; p < Pq; p += 16) cr[p] = __expf(lr[p] - m) * inv;
    }
}

// squash(s): v = (|s|^2/(1+|s|^2)) * s/|s|   (one thread per (b,o))
__device__ __forceinline__ void squash_pass(int tid,
                                            const float* __restrict__ s_red,
                                            float* __restrict__ v_lds,
                                            float* __restrict__ gout,
                                            int n, int b0)
{
    if (tid < 256) {
        int b = tid >> 4, o = tid & 15;
        float sv = s_red[b * SSTR + o];
        float sq = sv * sv;
        sq += __shfl_xor(sq, 1);
        sq += __shfl_xor(sq, 2);
        sq += __shfl_xor(sq, 4);
        sq += __shfl_xor(sq, 8);
        float scale = (sq / (1.0f + sq)) * rsqrtf(sq);
        float v = sv * scale;
        if (gout) gout[((size_t)n * Bq + (size_t)(b0 + b)) * COq + o] = v;
        else      v_lds[b * SSTR + o] = v;
    }
}

extern "C" __global__ void __launch_bounds__(NTHREADS)
digitcaps_routing_kernel(const float* __restrict__ x,
                         const float* __restrict__ W,
                         float* __restrict__ out)
{
    __shared__ float logits[16 * LSTR];   // routing logits b_ij (per b-row, per p)
    __shared__ float cw[16 * LSTR];       // coupling coefficients c_ij
    __shared__ float s_red[16 * SSTR];    // s_j reduction tile
    __shared__ float v_lds[16 * SSTR];    // v_j tile

    const int tid  = threadIdx.x;
    const int lane = tid & 31;
    const int wave = tid >> 5;
    const int hh   = (lane >> 4) & 1;
    const int mrow = lane & 15;

    const int n  = blockIdx.x / (Bq / 16);
    const int b0 = (blockIdx.x % (Bq / 16)) * 16;

    const float* xrow  = x + (size_t)(b0 + mrow) * Pq * CIq;   // this lane's A row
    const float* wbase = W + (size_t)n * Pq * CIq * COq;       // this block's W[n]

    const int p_begin = wave * PPW;
    const int p_end   = p_begin + PPW;

    // ---- iteration 0: c is uniform (softmax of zeros) ----
    for (int i = tid; i < 16 * SSTR; i += NTHREADS) s_red[i] = 0.f;
    __syncthreads();
    s_pass(xrow, wbase, p_begin, p_end, hh, mrow, cw, s_red, /*uniform=*/true);
    __syncthreads();
    squash_pass(tid, s_red, v_lds, nullptr, n, b0);
    __syncthreads();

    // ---- iterations 1..2 ----
    for (int it = 1; it < 3; ++it) {
        a_pass(xrow, wbase, p_begin, p_end, hh, mrow, v_lds, logits, /*accum=*/(it == 2));
        __syncthreads();
        stats_pass(tid, logits, cw);
        for (int i = tid; i < 16 * SSTR; i += NTHREADS) s_red[i] = 0.f;
        __syncthreads();
        s_pass(xrow, wbase, p_begin, p_end, hh, mrow, cw, s_red, /*uniform=*/false);
        __syncthreads();
        squash_pass(tid, s_red, v_lds, (it == 2) ? out : nullptr, n, b0);
        __syncthreads();
    }
}

extern "C" void kernel_launch(void* const* d_in, const int* in_sizes, int n_in,
                              void* d_out, int out_size, void* d_ws, size_t ws_size,
                              hipStream_t stream)
{
    (void)in_sizes; (void)n_in; (void)out_size; (void)d_ws; (void)ws_size;
    const float* x = (const float*)d_in[0];   // [512, 1152, 8]
    const float* W = (const float*)d_in[1];   // [10, 1152, 8, 16]
    float* out = (float*)d_out;               // [10, 512, 16]
    dim3 grid(Nq * (Bq / 16));                // 320 workgroups
    dim3 block(NTHREADS);                     // 16 waves (wave32)
    hipLaunchKernelGGL(digitcaps_routing_kernel, grid, block, 0, stream, x, W, out);
}